// Multihead_Latent_attention_Decoupled_rope_62268435857972
// MI455X (gfx1250) — compile-verified
//
#include <hip/hip_runtime.h>
#include <hip/hip_bf16.h>

// ---------------------------------------------------------------------------
// MLA with decoupled RoPE on gfx1250 (MI455X), fp32 path via
// V_WMMA_F32_16X16X4_F32 (exact fp32 matrix math; wave32, 16x16 tiles, K=4).
// GEMM B-operand staged through LDS with K-pair interleaving so each WMMA
// B-fragment is a single aligned ds_load_b64; global_prefetch for next chunk.
// ---------------------------------------------------------------------------

typedef __attribute__((ext_vector_type(2))) float v2f;
typedef __attribute__((ext_vector_type(8))) float v8f;

#define BATCH   4
#define SEQ     1024
#define NHEAD   16
#define DHEAD   64
#define TTOK    (BATCH * SEQ)        // 4096 tokens
#define DKV     512
#define ROPED   32                   // rope dims per head (cos width = 16)

#define GEMM_KB 32                   // K-chunk staged in LDS
#define PAIR_LD 160                  // floats per k-pair row (128 data + 32 pad):
                                     // stride % 64 == 32 -> half-waves on
                                     // disjoint bank windows for b64 reads

__device__ __forceinline__ v8f wmma4(v2f a, v2f b, v8f c) {
  // D = A(16x4,f32) * B(4x16,f32) + C(16x16,f32)
  return __builtin_amdgcn_wmma_f32_16x16x4_f32(
      /*neg_a=*/false, a, /*neg_b=*/false, b,
      /*c_mod=*/(short)0, c, /*reuse_a=*/false, /*reuse_b=*/false);
}

// ---------------------------------------------------------------------------
// Generic fp32 WMMA GEMM: C[M x N] = A[M x K] @ B[K x N] (+ bias[N]).
// Block = 128 threads (4 waves); wave w computes rows [blockIdx.y*64 + w*16, +16),
// cols [blockIdx.x*64, +64) as 4 16x16 WMMA tiles.
// B chunk (32 x 64) staged in LDS once per block (4x reuse, coalesced loads),
// K-pair interleaved: element (kk, c) lives at Bs[(kk>>1)*PAIR_LD + 2c + (kk&1)],
// so a B fragment {B[ka][c], B[ka+1][c]} is one aligned ds_load_b64.
// Column clamping + guarded stores handle N % 64 != 0 (N=544 case).
// ---------------------------------------------------------------------------
__global__ void gemm_wmma_f32(const float* __restrict__ A,
                              const float* __restrict__ Bm,
                              const float* __restrict__ bias,
                              float* __restrict__ C,
                              int N, int K, int lda, int ldb, int ldc) {
  __shared__ float Bs[(GEMM_KB / 2) * PAIR_LD];

  const int lane = threadIdx.x & 31;
  const int wave = threadIdx.x >> 5;
  const int row  = lane & 15;         // A: M index, B: N index, C: col index
  const int hi   = lane >> 4;         // K sub-offset selector
  const int m0   = (blockIdx.y * 4 + wave) * 16;
  const int n0   = blockIdx.x * 64;

  v8f acc0 = {}, acc1 = {}, acc2 = {}, acc3 = {};

  const float* arow = A + (size_t)(m0 + row) * lda;
  const int c0 = (0 * 16 + row) * 2;  // interleaved B column offsets
  const int c1 = (1 * 16 + row) * 2;
  const int c2 = (2 * 16 + row) * 2;
  const int c3 = (3 * 16 + row) * 2;

  for (int k0 = 0; k0 < K; k0 += GEMM_KB) {
    // Prefetch next B chunk: 128 threads cover 32 rows x 4 x 16-float segments.
    if (k0 + GEMM_KB < K) {
      const int pr = threadIdx.x >> 2;                 // 0..31 (row in chunk)
      const int pc = (threadIdx.x & 3) << 4;           // 0,16,32,48
      const int pn = min(n0 + pc, N - 1);
      __builtin_prefetch(&Bm[(size_t)(k0 + GEMM_KB + pr) * ldb + pn], 0, 3);
    }
    __syncthreads();
    // Stage B[k0..k0+31][n0..n0+63] -> LDS, pair-interleaved (coalesced reads;
    // clamped for ragged N).
    for (int i = threadIdx.x; i < GEMM_KB * 64; i += 128) {
      const int kk = i >> 6;
      const int cc = i & 63;
      const int nn = min(n0 + cc, N - 1);
      Bs[(kk >> 1) * PAIR_LD + cc * 2 + (kk & 1)] =
          Bm[(size_t)(k0 + kk) * ldb + nn];
    }
    __syncthreads();

#pragma unroll
    for (int kk = 0; kk < GEMM_KB; kk += 4) {
      v2f a = *(const v2f*)(arow + k0 + kk + hi * 2);  // A[m][..], aligned b64
      const float* bp = &Bs[((kk >> 1) + hi) * PAIR_LD];
      acc0 = wmma4(a, *(const v2f*)(bp + c0), acc0);   // single ds_load_b64 each
      acc1 = wmma4(a, *(const v2f*)(bp + c1), acc1);
      acc2 = wmma4(a, *(const v2f*)(bp + c2), acc2);
      acc3 = wmma4(a, *(const v2f*)(bp + c3), acc3);
    }
  }

#pragma unroll
  for (int v = 0; v < 8; ++v) {
    const int m = m0 + v + 8 * hi;                    // C row per ISA layout
    float* crow = C + (size_t)m * ldc;
    int nn;
    nn = n0 + 0 * 16 + row; if (nn < N) crow[nn] = acc0[v] + (bias ? bias[nn] : 0.f);
    nn = n0 + 1 * 16 + row; if (nn < N) crow[nn] = acc1[v] + (bias ? bias[nn] : 0.f);
    nn = n0 + 2 * 16 + row; if (nn < N) crow[nn] = acc2[v] + (bias ? bias[nn] : 0.f);
    nn = n0 + 3 * 16 + row; if (nn < N) crow[nn] = acc3[v] + (bias ? bias[nn] : 0.f);
  }
}

// ---------------------------------------------------------------------------
// Row LayerNorm (jnp.var semantics, ddof=0), one 256-thread block per row.
// Safe in-place (each element re-read/written by the same thread).
// ---------------------------------------------------------------------------
__global__ void layernorm_rows(const float* __restrict__ X,
                               const float* __restrict__ g,
                               const float* __restrict__ bta,
                               float* __restrict__ Y,
                               int W, int ldx, int ldy) {
  const int r = blockIdx.x;
  const float* x = X + (size_t)r * ldx;
  float* y = Y + (size_t)r * ldy;

  float s = 0.f, sq = 0.f;
  for (int i = threadIdx.x; i < W; i += blockDim.x) {
    float v = x[i]; s += v; sq += v * v;
  }
#pragma unroll
  for (int off = 16; off > 0; off >>= 1) {
    s  += __shfl_xor(s, off);
    sq += __shfl_xor(sq, off);
  }
  __shared__ float red[2][8];
  const int wid = threadIdx.x >> 5, lane = threadIdx.x & 31;
  if (lane == 0) { red[0][wid] = s; red[1][wid] = sq; }
  __syncthreads();
  if (threadIdx.x == 0) {
    float a = 0.f, b2 = 0.f;
    for (int i = 0; i < 8; ++i) { a += red[0][i]; b2 += red[1][i]; }
    red[0][0] = a; red[1][0] = b2;
  }
  __syncthreads();
  const float mean = red[0][0] / (float)W;
  const float var  = red[1][0] / (float)W - mean * mean;
  const float inv  = rsqrtf(var + 1e-5f);
  for (int i = threadIdx.x; i < W; i += blockDim.x)
    y[i] = (x[i] - mean) * inv * g[i] + bta[i];
}

// ---------------------------------------------------------------------------
// Decoupled RoPE: cos table width 16, inv_freq[j] = 10000^{-j/8}, t = s/40.
// Applied to first 16 of the 32 rope dims; last 16 pass through.
// ---------------------------------------------------------------------------
__device__ __forceinline__ float rope_val(const float* __restrict__ xr, int dr, int s) {
  if (dr >= 16) return xr[dr];
  const int j = dr & 7;
  const float ang = ((float)s * (1.0f / 40.0f)) * __powf(10000.0f, -(float)j * 0.125f);
  const float c = __cosf(ang), sn = __sinf(ang);
  const float rot = (dr < 8) ? -xr[dr + 8] : xr[dr - 8];
  return xr[dr] * c + rot * sn;
}

// q[b][h][s][d] : d<32 from Q (nope), d>=32 rope(Qr)
__global__ void pack_q(const float* __restrict__ Q, const float* __restrict__ Qr,
                       float* __restrict__ qout) {
  const int idx = blockIdx.x * blockDim.x + threadIdx.x;   // TTOK*NHEAD*64
  if (idx >= TTOK * NHEAD * 64) return;
  const int d = idx & 63;
  const int h = (idx >> 6) & 15;
  const int t = idx >> 10;
  const int s = t & (SEQ - 1);
  const int b = t >> 10;
  float val;
  if (d < 32) val = Q[(size_t)t * 512 + h * 32 + d];
  else        val = rope_val(Qr + (size_t)t * 512 + h * 32, d - 32, s);
  qout[(((size_t)(b * NHEAD + h)) * SEQ + s) * 64 + d] = val;
}

// k[b][h][s][d] (d<32 = K-nope, d>=32 = rope(Kr) broadcast over heads),
// v[b][h][s][dv] from KV[..., 32:96]
__global__ void pack_kv(const float* __restrict__ KV, const float* __restrict__ ckv,
                        float* __restrict__ kout, float* __restrict__ vout) {
  const int idx = blockIdx.x * blockDim.x + threadIdx.x;   // TTOK*NHEAD*128
  if (idx >= TTOK * NHEAD * 128) return;
  const int d = idx & 127;
  const int h = (idx >> 7) & 15;
  const int t = idx >> 11;
  const int s = t & (SEQ - 1);
  const int b = t >> 10;
  const size_t base = (((size_t)(b * NHEAD + h)) * SEQ + s) * 64;
  if (d < 64) {
    float val;
    if (d < 32) val = KV[(size_t)t * 1536 + h * 96 + d];
    else        val = rope_val(ckv + (size_t)t * 544 + 512, d - 32, s);
    kout[base + d] = val;
  } else {
    const int dv = d - 64;
    vout[base + dv] = KV[(size_t)t * 1536 + h * 96 + 32 + dv];
  }
}

// ---------------------------------------------------------------------------
// Causal attention, one workgroup (4 waves) per 16-row query block per (b,h).
// LDS: 16x1024 fp32 score/P tile (64 KB of the WGP's 320 KB).
//  Phase 1: S = (Q K^T) * 1/8 via WMMA (only causal-live key blocks).
//  Phase 2: exact causal softmax, P normalized in LDS (8 lanes per row).
//  Phase 3: O = P V via WMMA, A-fragments from LDS, output to (B,S,H*64).
// ---------------------------------------------------------------------------
__global__ void mla_attention(const float* __restrict__ q, const float* __restrict__ k,
                              const float* __restrict__ v, float* __restrict__ out) {
  extern __shared__ float Sc[];                        // [16][1024]
  const int qb = blockIdx.x;                           // query block 0..63
  const int bh = blockIdx.y;                           // 0..63
  const int b = bh >> 4, h = bh & 15;
  const int q0 = qb * 16;
  const int lane = threadIdx.x & 31, wave = threadIdx.x >> 5;
  const int row = lane & 15, hi = lane >> 4;

  const float* qp = q + (size_t)bh * SEQ * 64;
  const float* kp = k + (size_t)bh * SEQ * 64;
  const float* vp = v + (size_t)bh * SEQ * 64;

  // Q tile A-fragments (16 rows x 64 dims), reused across all key blocks.
  v2f aq[16];
#pragma unroll
  for (int ks = 0; ks < 16; ++ks)
    aq[ks] = *(const v2f*)(qp + (size_t)(q0 + row) * 64 + ks * 4 + hi * 2);

  const float scale = 0.125f;                          // 1/sqrt(64)
  for (int kb = wave; kb <= qb; kb += 4) {             // causal-live key blocks
    v8f acc = {};
#pragma unroll
    for (int ks = 0; ks < 16; ++ks) {
      v2f bb = *(const v2f*)(kp + (size_t)(kb * 16 + row) * 64 + ks * 4 + hi * 2);
      acc = wmma4(aq[ks], bb, acc);
    }
#pragma unroll
    for (int vv = 0; vv < 8; ++vv)
      Sc[(vv + 8 * hi) * 1024 + kb * 16 + row] = acc[vv] * scale;
  }
  __syncthreads();

  // Softmax with exact causal mask; P (normalized) left in LDS.
  {
    const int r = threadIdx.x >> 3;                    // 0..15
    const int sub = threadIdx.x & 7;
    const int lastcol = q0 + r;                        // inclusive causal bound
    float m = -3.0e38f;
    for (int t = sub; t <= lastcol; t += 8) m = fmaxf(m, Sc[r * 1024 + t]);
#pragma unroll
    for (int off = 1; off < 8; off <<= 1) m = fmaxf(m, __shfl_xor(m, off));
    float sum = 0.f;
    for (int t = sub; t < SEQ; t += 8) {
      const float p = (t <= lastcol) ? __expf(Sc[r * 1024 + t] - m) : 0.f;
      Sc[r * 1024 + t] = p;
      sum += p;
    }
#pragma unroll
    for (int off = 1; off < 8; off <<= 1) sum += __shfl_xor(sum, off);
    const float inv = 1.0f / sum;
    for (int t = sub; t <= lastcol; t += 8) Sc[r * 1024 + t] *= inv;
  }
  __syncthreads();

  // O = P @ V ; wave w owns output cols [w*16, w*16+16).
  v8f o = {};
  const int col = wave * 16 + row;
  const int ksmax = (qb + 1) * 4;                      // P is zero beyond causal bound
  for (int ks = 0; ks < ksmax; ++ks) {
    const int kd = ks * 4 + hi * 2;
    v2f ap = *(const v2f*)(Sc + row * 1024 + kd);
    v2f bv = (v2f){vp[(size_t)kd * 64 + col], vp[(size_t)(kd + 1) * 64 + col]};
    o = wmma4(ap, bv, o);
  }
#pragma unroll
  for (int vv = 0; vv < 8; ++vv) {
    const int r = vv + 8 * hi;
    out[((size_t)(b * SEQ) + q0 + r) * 1024 + h * 64 + col] = o[vv];
  }
}

// ---------------------------------------------------------------------------
// Host-side orchestration
// ---------------------------------------------------------------------------
extern "C" void kernel_launch(void* const* d_in, const int* in_sizes, int n_in,
                              void* d_out, int out_size, void* d_ws, size_t ws_size,
                              hipStream_t stream) {
  (void)in_sizes; (void)n_in; (void)out_size; (void)ws_size;

  const float* x     = (const float*)d_in[0];
  /* d_in[1] = mask (int32 tril) — causal mask applied analytically */
  const float* W_dq  = (const float*)d_in[2];
  const float* gq    = (const float*)d_in[3];
  const float* bq    = (const float*)d_in[4];
  const float* W_uq  = (const float*)d_in[5];
  const float* W_qr  = (const float*)d_in[6];
  const float* W_dkv = (const float*)d_in[7];
  const float* gkv   = (const float*)d_in[8];
  const float* bkv   = (const float*)d_in[9];
  const float* W_ukv = (const float*)d_in[10];
  const float* b_ukv = (const float*)d_in[11];
  const float* W0    = (const float*)d_in[12];

  float* out1 = (float*)d_out;                       // (B,S,1024)
  float* ckv  = out1 + (size_t)TTOK * 1024;          // (B,S,544) raw (pre-LN)

  // Workspace carve-up (fp32):
  float* ws = (float*)d_ws;
  float* c_q    = ws;  ws += (size_t)TTOK * 512;     // LN'd q-lora
  float* kvl    = ws;  ws += (size_t)TTOK * 512;     // LN'd kv-lora
  float* attn_o = ws;                                 // (B,S,1024), aliases Qbuf+Qrbuf
  float* Qbuf   = ws;  ws += (size_t)TTOK * 512;     // Q nope (consumed by pack_q)
  float* Qrbuf  = ws;  ws += (size_t)TTOK * 512;     // Q rope (consumed by pack_q)
  float* KVbuf  = ws;  ws += (size_t)TTOK * 1536;    // KV up-projection
  float* qws    = ws;  ws += (size_t)BATCH * NHEAD * SEQ * 64;
  float* kws    = ws;  ws += (size_t)BATCH * NHEAD * SEQ * 64;
  float* vws    = ws;  ws += (size_t)BATCH * NHEAD * SEQ * 64;

  const dim3 gblk(128);

  // 1) c_q = LN(x @ W_dq)
  gemm_wmma_f32<<<dim3(512 / 64, TTOK / 64), gblk, 0, stream>>>(
      x, W_dq, nullptr, c_q, 512, 1024, 1024, 512, 512);
  layernorm_rows<<<TTOK, 256, 0, stream>>>(c_q, gq, bq, c_q, 512, 512, 512);

  // 2) ckv = x @ W_dkv (raw -> output #2); kv_lora = LN(ckv[:, :512])
  gemm_wmma_f32<<<dim3((544 + 63) / 64, TTOK / 64), gblk, 0, stream>>>(
      x, W_dkv, nullptr, ckv, 544, 1024, 1024, 544, 544);
  layernorm_rows<<<TTOK, 256, 0, stream>>>(ckv, gkv, bkv, kvl, 512, 544, 512);

  // 3) Q = c_q @ W_uq ; Qr = c_q @ W_qr ; RoPE-pack into (B,H,S,64)
  gemm_wmma_f32<<<dim3(512 / 64, TTOK / 64), gblk, 0, stream>>>(
      c_q, W_uq, nullptr, Qbuf, 512, 512, 512, 512, 512);
  gemm_wmma_f32<<<dim3(512 / 64, TTOK / 64), gblk, 0, stream>>>(
      c_q, W_qr, nullptr, Qrbuf, 512, 512, 512, 512, 512);
  pack_q<<<(TTOK * NHEAD * 64) / 256, 256, 0, stream>>>(Qbuf, Qrbuf, qws);

  // 4) KV = kv_lora @ W_ukv + b_ukv ; pack K (nope+rope) and V into (B,H,S,64)
  gemm_wmma_f32<<<dim3(1536 / 64, TTOK / 64), gblk, 0, stream>>>(
      kvl, W_ukv, b_ukv, KVbuf, 1536, 512, 512, 1536, 1536);
  pack_kv<<<(TTOK * NHEAD * 128) / 256, 256, 0, stream>>>(KVbuf, ckv, kws, vws);

  // 5) Causal attention -> attn_o (B,S,H*64); 64 KB dynamic LDS per workgroup
  mla_attention<<<dim3(SEQ / 16, BATCH * NHEAD), 128, 16 * 1024 * sizeof(float), stream>>>(
      qws, kws, vws, attn_o);

  // 6) out = attn_o @ W0 -> output #1
  gemm_wmma_f32<<<dim3(1024 / 64, TTOK / 64), gblk, 0, stream>>>(
      attn_o, W0, nullptr, out1, 1024, 1024, 1024, 1024, 1024);
}